// LearnableEdgeMaskplus_45921790329387
// MI455X (gfx1250) — compile-verified
//
#include <hip/hip_runtime.h>
#include <hip/hip_bf16.h>

#define N_NODES 100000
#define N_EDGES 50000
#define NNZ     1000000
#define DIM     128
#define HID     32
#define KSEL    25000

typedef float v2f __attribute__((ext_vector_type(2)));
typedef float v8f __attribute__((ext_vector_type(8)));

// ---------------------------------------------------------------- zero ws
__global__ __launch_bounds__(256) void zero_kernel(float* p, int n) {
    int i = blockIdx.x * 256 + threadIdx.x;
    if (i < n) p[i] = 0.0f;
}

// ---------------------------------------------------------- scatter mean
// one wave32 per incidence pair: lane loads float4 of x[v], atomically adds
// into sums[e].  lane0 bumps the edge count.
__global__ __launch_bounds__(256) void scatter_kernel(const float* __restrict__ x,
                                                      const int* __restrict__ V,
                                                      const int* __restrict__ Ei,
                                                      float* __restrict__ sums,
                                                      float* __restrict__ cnt,
                                                      int nnz) {
    int pair = blockIdx.x * 8 + (threadIdx.x >> 5);
    int lane = threadIdx.x & 31;
    if (pair >= nnz) return;
    int v = V[pair];
    int e = Ei[pair];
    float4 xv = reinterpret_cast<const float4*>(x)[(size_t)v * 32 + lane];
    float* dst = sums + (size_t)e * DIM + lane * 4;
    unsafeAtomicAdd(dst + 0, xv.x);
    unsafeAtomicAdd(dst + 1, xv.y);
    unsafeAtomicAdd(dst + 2, xv.z);
    unsafeAtomicAdd(dst + 3, xv.w);
    if (lane == 0) unsafeAtomicAdd(cnt + e, 1.0f);
}

// ------------------------------------------------------------ WMMA GEMM
// h[E,32] = relu( (sums/max(cnt,1)) @ W1^T + b1 )
// block = 256 threads = 8 waves; each wave owns one 16-edge tile and two
// f32 16x16 accumulators (N=0..15, N=16..31).  K loop: 32 x V_WMMA_F32_16X16X4_F32.
__global__ __launch_bounds__(256) void gemm_kernel(const float* __restrict__ sums,
                                                   const float* __restrict__ cnt,
                                                   const float* __restrict__ W1,
                                                   const float* __restrict__ b1,
                                                   float* __restrict__ h,
                                                   int E) {
    __shared__ float W1s[HID * 129];          // padded rows: stride 129
    __shared__ float AsAll[8 * 16 * 132];     // per-wave A tile, stride 132

    int wave = threadIdx.x >> 5;
    int lane = threadIdx.x & 31;

    // cooperative load of W1 [32,128] row-major into padded LDS
    for (int i = threadIdx.x; i < HID * DIM; i += 256) {
        W1s[(i >> 7) * 129 + (i & 127)] = W1[i];
    }

    int tile = blockIdx.x * 8 + wave;
    int base = tile * 16;
    float* As = &AsAll[wave * 16 * 132];

    // stage A tile (16 rows x 128 cols), applying the segment-mean division
    for (int it = 0; it < 16; ++it) {
        int linear = it * 32 + lane;
        int row = linear >> 5;      // 0..15
        int c4  = linear & 31;      // float4 column
        int g = base + row;
        float4 val = make_float4(0.f, 0.f, 0.f, 0.f);
        if (g < E) {
            val = reinterpret_cast<const float4*>(sums)[(size_t)g * 32 + c4];
            float inv = 1.0f / fmaxf(cnt[g], 1.0f);
            val.x *= inv; val.y *= inv; val.z *= inv; val.w *= inv;
        }
        float* dst = &As[row * 132 + c4 * 4];
        dst[0] = val.x; dst[1] = val.y; dst[2] = val.z; dst[3] = val.w;
    }
    __syncthreads();

    v8f c0 = {};
    v8f c1 = {};
    int m = lane & 15;
    int n = lane & 15;
    int khalf = (lane >> 4) * 2;   // 16x4 f32 A/B layout: lanes 16-31 hold K+2..K+3

    for (int k0 = 0; k0 < DIM; k0 += 4) {
        int kk = k0 + khalf;
        v2f a;  a.x  = As[m * 132 + kk];          a.y  = As[m * 132 + kk + 1];
        v2f bA; bA.x = W1s[n * 129 + kk];         bA.y = W1s[n * 129 + kk + 1];
        v2f bB; bB.x = W1s[(n + 16) * 129 + kk];  bB.y = W1s[(n + 16) * 129 + kk + 1];
        c0 = __builtin_amdgcn_wmma_f32_16x16x4_f32(false, a, false, bA, (short)0, c0, false, false);
        c1 = __builtin_amdgcn_wmma_f32_16x16x4_f32(false, a, false, bB, (short)0, c1, false, false);
    }

    // C/D layout: VGPR j -> M=j (lanes 0-15), M=j+8 (lanes 16-31); N = lane&15
    int rbase = base + ((lane >> 4) ? 8 : 0);
    int col = lane & 15;
    #pragma unroll
    for (int j = 0; j < 8; ++j) {
        int row = rbase + j;
        if (row < E) {
            float v0 = c0[j] + b1[col];
            float v1 = c1[j] + b1[col + 16];
            h[(size_t)row * HID + col]      = fmaxf(v0, 0.0f);
            h[(size_t)row * HID + col + 16] = fmaxf(v1, 0.0f);
        }
    }
}

// ------------------------------------------------- logits + sigmoid
__global__ __launch_bounds__(256) void logits_kernel(const float* __restrict__ h,
                                                     const float* __restrict__ W2,
                                                     const float* __restrict__ b2,
                                                     float* __restrict__ out,
                                                     int E) {
    int e = blockIdx.x * 256 + threadIdx.x;
    if (e >= E) return;
    float acc = b2[0];
    const float* hr = h + (size_t)e * HID;
    #pragma unroll
    for (int j = 0; j < HID; ++j) acc += hr[j] * W2[j];
    out[e] = 1.0f / (1.0f + __expf(-acc));
}

// ------------------------------------------------- deterministic top-K
// single block: 4-pass radix select on sign-flipped float keys, then an
// index-ordered tie-break scan (matches jax.lax.top_k stability).
__global__ __launch_bounds__(1024) void topk_kernel(const float* __restrict__ probs,
                                                    float* __restrict__ out,
                                                    int E, int K) {
    __shared__ unsigned hist[256];
    __shared__ unsigned scanbuf[1024];
    __shared__ unsigned running;

    int t = threadIdx.x;
    unsigned prefix = 0;
    int Kp = K;

    for (int pass = 0; pass < 4; ++pass) {
        int shift = 24 - 8 * pass;
        if (t < 256) hist[t] = 0;
        __syncthreads();
        for (int i = t; i < E; i += 1024) {
            unsigned u = __float_as_uint(probs[i]);
            u ^= (u >> 31) ? 0xFFFFFFFFu : 0x80000000u;
            if (pass == 0 || (u >> (shift + 8)) == prefix) {
                atomicAdd(&hist[(u >> shift) & 0xFFu], 1u);
            }
        }
        __syncthreads();
        unsigned cum = 0, chosen = 0;
        for (int b = 255; b >= 0; --b) {
            unsigned c = hist[b];
            if (cum + c >= (unsigned)Kp) { chosen = (unsigned)b; break; }
            cum += c;
        }
        prefix = (prefix << 8) | chosen;
        Kp -= (int)cum;
        __syncthreads();
    }
    unsigned thresh = prefix;   // K-th largest key; Kp ties to accept (lowest idx)

    if (t == 0) running = 0;
    __syncthreads();

    int Epad = ((E + 1023) / 1024) * 1024;
    for (int base = 0; base < Epad; base += 1024) {
        int i = base + t;
        unsigned u = 0; float p = 0.f; int valid = (i < E);
        if (valid) {
            p = probs[i];
            u = __float_as_uint(p);
            u ^= (u >> 31) ? 0xFFFFFFFFu : 0x80000000u;
        }
        unsigned tie = (valid && u == thresh) ? 1u : 0u;
        scanbuf[t] = tie;
        __syncthreads();
        for (int off = 1; off < 1024; off <<= 1) {
            unsigned v = (t >= off) ? scanbuf[t - off] : 0u;
            __syncthreads();
            scanbuf[t] += v;
            __syncthreads();
        }
        unsigned rank = running + (scanbuf[t] - tie);
        if (valid) {
            float hard = (u > thresh) ? 1.0f : ((tie && rank < (unsigned)Kp) ? 1.0f : 0.0f);
            float soft = (hard - p) + p;         // straight-through, as written
            out[(size_t)E + i]     = soft;
            out[(size_t)2 * E + i] = hard;
        }
        __syncthreads();
        if (t == 0) running += scanbuf[1023];
        __syncthreads();
    }
}

// ------------------------------------------------------------------ launch
extern "C" void kernel_launch(void* const* d_in, const int* in_sizes, int n_in,
                              void* d_out, int out_size, void* d_ws, size_t ws_size,
                              hipStream_t stream) {
    const float* x  = (const float*)d_in[0];
    const int*   V  = (const int*)d_in[1];
    const int*   Ei = (const int*)d_in[2];
    const float* W1 = (const float*)d_in[3];
    const float* b1 = (const float*)d_in[4];
    const float* W2 = (const float*)d_in[5];
    const float* b2 = (const float*)d_in[6];
    float* out = (float*)d_out;

    float* sums = (float*)d_ws;                       // [E,128]
    float* cnt  = sums + (size_t)N_EDGES * DIM;       // [E]
    float* h    = cnt + N_EDGES;                      // [E,32]

    int nz = N_EDGES * DIM + N_EDGES;
    zero_kernel<<<(nz + 255) / 256, 256, 0, stream>>>(sums, nz);

    scatter_kernel<<<(NNZ + 7) / 8, 256, 0, stream>>>(x, V, Ei, sums, cnt, NNZ);

    int numTiles = (N_EDGES + 15) / 16;               // 3125
    gemm_kernel<<<(numTiles + 7) / 8, 256, 0, stream>>>(sums, cnt, W1, b1, h, N_EDGES);

    logits_kernel<<<(N_EDGES + 255) / 256, 256, 0, stream>>>(h, W2, b2, out, N_EDGES);

    topk_kernel<<<1, 1024, 0, stream>>>(out, out, N_EDGES, KSEL);
}